// GCN_20280835571967
// MI455X (gfx1250) — compile-verified
//
#include <hip/hip_runtime.h>
#include <hip/hip_bf16.h>

// GCN 2-layer forward for MI455X (gfx1250, wave32).
// fp32 end-to-end (memory/L2-bound workload; the ~58MB working set is L2-resident).
// Dense projections use V_WMMA_F32_16X16X4_F32 (16x16 tile per wave, K-step 4).
// Segment-sum scatters use hardware fp32 global atomics (global_atomic_add_f32,
// non-returning, device scope) via inline asm to guarantee no CAS expansion.

#define IN_DIM  64
#define OUT_DIM 16

typedef __attribute__((ext_vector_type(2))) float v2f;
typedef __attribute__((ext_vector_type(8))) float v8f;

// Guaranteed-hardware fp32 atomic add: VGLOBAL GLOBAL_ATOMIC_ADD_F32 (opcode 86),
// non-returning (STOREcnt-tracked), device scope so the RMW resolves at L2.
__device__ __forceinline__ void atomic_add_f32(float* p, float v) {
  asm volatile("global_atomic_add_f32 %0, %1, off scope:SCOPE_DEV"
               :: "v"(p), "v"(v) : "memory");
}

// ---------------- utility kernels ----------------

__global__ void fill0_kernel(float* __restrict__ p, int n) {
  int i = blockIdx.x * blockDim.x + threadIdx.x;
  if (i < n) p[i] = 0.0f;
}

__global__ void degree_kernel(const int* __restrict__ dst, float* __restrict__ deg, int E) {
  int i = blockIdx.x * blockDim.x + threadIdx.x;
  if (i < E) atomic_add_f32(&deg[dst[i]], 1.0f);
}

// dinv = rsqrt(deg + 1)  (+1 accounts for the self-loop at every node)
__global__ void rsqrt_kernel(float* __restrict__ d, int n) {
  int i = blockIdx.x * blockDim.x + threadIdx.x;
  if (i < n) d[i] = rsqrtf(d[i] + 1.0f);
}

__global__ void bias_relu_kernel(float* __restrict__ a, const float* __restrict__ b, int n) {
  int i = blockIdx.x * blockDim.x + threadIdx.x;
  if (i < n) a[i] = fmaxf(a[i] + b[i & 63], 0.0f);
}

__global__ void bias_kernel(float* __restrict__ a, const float* __restrict__ b, int n) {
  int i = blockIdx.x * blockDim.x + threadIdx.x;
  if (i < n) a[i] = a[i] + b[i & 15];
}

// ---------------- WMMA GEMM: out[M,NCOLS] = A[M,64] * W[64,NCOLS] ----------------
// One 16x16 output tile per wave; 8 waves (256 threads) per block.
// W is staged into LDS once per block and read as B-fragments (ds_load).
//
// Fragment layouts (ISA 7.12.2, wave32):
//   A (16x4 f32): lanes 0-15 -> rows, VGPR0/1 hold K = kh, kh+1 with kh = (lane>>4)*2
//   B (4x16 f32): lanes 0-15 -> cols, VGPR0/1 hold K = kh, kh+1 (row-striped, mirrored)
//   C/D (16x16 f32): VGPR r, lanes 0-15 -> (M=r, N=lane); lanes 16-31 -> (M=r+8, N=lane-16)

template <int NCOLS>
__global__ __launch_bounds__(256) void gemm_wmma_kernel(const float* __restrict__ A,
                                                        const float* __restrict__ W,
                                                        float* __restrict__ out, int M) {
  __shared__ float sW[64 * NCOLS];
  const int nvec = (64 * NCOLS) / 4;
  for (int i = threadIdx.x; i < nvec; i += blockDim.x)
    ((float4*)sW)[i] = ((const float4*)W)[i];
  __syncthreads();

  const int lane = threadIdx.x & 31;
  const int wid  = threadIdx.x >> 5;
  const int ntile_n = NCOLS / 16;
  const int mtiles  = (M + 15) >> 4;
  const int gt = blockIdx.x * 8 + wid;
  const int mt = gt / ntile_n;
  const int nt = gt - mt * ntile_n;
  if (mt >= mtiles) return;

  const int m0 = mt * 16;
  const int n0 = nt * 16;
  const int row = lane & 15;
  const int kh  = (lane >> 4) << 1;                 // 0 or 2
  const int arow_idx = min(m0 + row, M - 1);        // clamp for ragged M (exact here)
  const float* __restrict__ arow = A + (size_t)arow_idx * 64 + kh;
  const float* __restrict__ bcol = sW + n0 + row;   // row == lane&15 doubles as col for B

  v8f c = {};
#pragma unroll
  for (int k = 0; k < 64; k += 4) {
    v2f a, b;
    a.x = arow[k];
    a.y = arow[k + 1];
    b.x = bcol[(k + kh) * NCOLS];
    b.y = bcol[(k + kh + 1) * NCOLS];
    c = __builtin_amdgcn_wmma_f32_16x16x4_f32(false, a, false, b, (short)0, c, false, false);
  }

  const int rbase = (lane >> 4) << 3;               // 0 or 8
#pragma unroll
  for (int r = 0; r < 8; ++r) {
    int orow = m0 + rbase + r;
    if (orow < M) out[(size_t)orow * NCOLS + n0 + row] = c[r];
  }
}

// ---------------- edge scatter (segment-sum with symmetric normalization) ----------------

// Layer 1: 64 columns. One wave per (virtual) edge; lane handles 2 columns (float2).
__global__ __launch_bounds__(256) void scatter64_kernel(const float* __restrict__ h,
                                                        const int* __restrict__ src,
                                                        const int* __restrict__ dst,
                                                        const float* __restrict__ dinv,
                                                        float* __restrict__ agg,
                                                        int E, int Nn) {
  const int e = blockIdx.x * 8 + (threadIdx.x >> 5);
  const int lane = threadIdx.x & 31;
  if (e >= E + Nn) return;
  int s, d;
  if (e < E) { s = src[e]; d = dst[e]; } else { s = e - E; d = s; }
  const float w = dinv[s] * dinv[d];
  const float2 v = ((const float2*)(h + (size_t)s * 64))[lane];
  float* op = agg + (size_t)d * 64 + lane * 2;
  atomic_add_f32(op,     v.x * w);
  atomic_add_f32(op + 1, v.y * w);
}

// Layer 2: 16 columns. One thread per (edge, column); consecutive threads hit
// consecutive columns of the same destination row (coalesced atomics).
__global__ __launch_bounds__(256) void scatter16_kernel(const float* __restrict__ h,
                                                        const int* __restrict__ src,
                                                        const int* __restrict__ dst,
                                                        const float* __restrict__ dinv,
                                                        float* __restrict__ out,
                                                        int E, int Nn) {
  const int tid = blockIdx.x * blockDim.x + threadIdx.x;
  const int e = tid >> 4;
  const int c = tid & 15;
  if (e >= E + Nn) return;
  int s, d;
  if (e < E) { s = src[e]; d = dst[e]; } else { s = e - E; d = s; }
  const float w = dinv[s] * dinv[d];
  atomic_add_f32(&out[(size_t)d * 16 + c], h[(size_t)s * 16 + c] * w);
}

// ---------------- launch ----------------

extern "C" void kernel_launch(void* const* d_in, const int* in_sizes, int n_in,
                              void* d_out, int out_size, void* d_ws, size_t ws_size,
                              hipStream_t stream) {
  const float* x    = (const float*)d_in[0];
  const int*   edge = (const int*)d_in[1];   // [2, E] row-major, int per harness convention
  const float* W1   = (const float*)d_in[2];
  const float* b1   = (const float*)d_in[3];
  const float* W2   = (const float*)d_in[4];
  const float* b2   = (const float*)d_in[5];
  float* out = (float*)d_out;

  const int E  = in_sizes[1] / 2;
  const int Nn = in_sizes[0] / IN_DIM;
  const int* src = edge;
  const int* dst = edge + E;

  // workspace layout (bytes): dinv[Nn] | h1[Nn*64] | agg1[Nn*64]; h2 aliases h1.
  char* ws = (char*)d_ws;
  float* dinv = (float*)ws;
  float* h1   = (float*)(ws + (size_t)Nn * 4);
  float* agg1 = (float*)(ws + (size_t)Nn * 4 + (size_t)Nn * 64 * 4);
  float* h2   = h1;   // h1 dead after scatter64; reuse for layer-2 projection

  const int TB = 256;
  const int mtiles = (Nn + 15) >> 4;

  // zero accumulators (harness poisons buffers; re-zero every call for determinism)
  fill0_kernel<<<(Nn + TB - 1) / TB, TB, 0, stream>>>(dinv, Nn);
  fill0_kernel<<<(Nn * 64 + TB - 1) / TB, TB, 0, stream>>>(agg1, Nn * 64);
  fill0_kernel<<<(Nn * 16 + TB - 1) / TB, TB, 0, stream>>>(out, Nn * 16);

  // symmetric degree normalization
  degree_kernel<<<(E + TB - 1) / TB, TB, 0, stream>>>(dst, dinv, E);
  rsqrt_kernel<<<(Nn + TB - 1) / TB, TB, 0, stream>>>(dinv, Nn);

  // layer 1: h1 = x @ W1 ; agg1 = segsum(norm * h1[src]) ; relu(agg1 + b1)
  gemm_wmma_kernel<64><<<(mtiles * 4 + 7) / 8, 256, 0, stream>>>(x, W1, h1, Nn);
  scatter64_kernel<<<((E + Nn) + 7) / 8, 256, 0, stream>>>(h1, src, dst, dinv, agg1, E, Nn);
  bias_relu_kernel<<<(Nn * 64 + TB - 1) / TB, TB, 0, stream>>>(agg1, b1, Nn * 64);

  // layer 2: h2 = agg1 @ W2 ; out = segsum(norm * h2[src]) + b2
  gemm_wmma_kernel<16><<<(mtiles + 7) / 8, 256, 0, stream>>>(agg1, W2, h2, Nn);
  scatter16_kernel<<<((E + Nn) * 16 + TB - 1) / TB, TB, 0, stream>>>(h2, src, dst, dinv, out, E, Nn);
  bias_kernel<<<(Nn * 16 + TB - 1) / TB, TB, 0, stream>>>(out, b2, Nn * 16);

  (void)n_in; (void)out_size; (void)ws_size;
}